// SparseGATLayerTemporal_88064009437902
// MI455X (gfx1250) — compile-verified
//
#include <hip/hip_runtime.h>
#include <hip/hip_bf16.h>

// ---------------------------------------------------------------------------
// SparseGATLayerTemporal for MI455X (gfx1250, wave32)
//
// Reference math:
//   h = (x * exp(-0.1*arange(128))) @ W          // 100000x128 @ 128x64 fp32
//   top-8 of 32 edge weights per node            // top_k(ew, 8)
//   out[n] = sum_k topw[n,k] * h[nbr[n,k], :]    // softmax over size-1 axis == 1,
//                                                // so attention == topw exactly.
//
// MI455X reasoning: total HBM traffic ~150 MB -> ~6.5us at 23.3 TB/s. The
// h-gather is served by the 192 MB L2 (h = 25.6 MB). GEMM is 1.6 GFLOP fp32
// -> V_WMMA_F32_16X16X4_F32, fp32 end-to-end (no precision downgrade needed;
// we're bandwidth-bound, not compute-bound).
//
// Round-2 refinement: one wave now produces the FULL 16x64 output stripe
// (4 N-tiles), reusing the A operand in registers across 4 WMMAs per k-step.
// This makes x a single 51.2 MB stream (was 4x re-read through L2) and gives
// a 4:1 wmma:A-load ratio in the inner loop.
// ---------------------------------------------------------------------------

typedef __attribute__((ext_vector_type(2))) float v2f;
typedef __attribute__((ext_vector_type(8))) float v8f;

#define N_NODES 100000
#define T_DIM   128
#define F_DIM   64
#define DEG     32
#define TOPK    8
#define LAM     0.1f

// ---------------------------------------------------------------------------
// Kernel 1: h = (x * decays) @ W via v_wmma_f32_16x16x4_f32.
// One wave per 16-row stripe of h (all 4 N-tiles). 10 waves (320 thr)/block:
// 6250 M-tiles = 625 blocks * 10 waves exactly.
// A-matrix f32 16x4 layout (ISA 7.12.2): lanes 0-15 hold rows M=0..15 with
// (K, K+1) in (v0, v1); lanes 16-31 hold (K+2, K+3).
// B-matrix 4x16: v0 = B[K+koff][n], v1 = B[K+koff+1][n], n = lane&15.
// C/D f32 16x16: VGPR r -> row M = r + (lane>=16 ? 8 : 0), col N = lane&15.
// ---------------------------------------------------------------------------
__global__ void __launch_bounds__(320)
gat_gemm_h_kernel(const float* __restrict__ x,
                  const float* __restrict__ W,
                  float* __restrict__ h)
{
    __shared__ float sW[T_DIM * F_DIM];   // 32 KB: full 128x64 W
    __shared__ float sDec[T_DIM];         // exp(-0.1*k)

    const int tid = threadIdx.x;

    // Stage W into LDS (row-major, same as global) and build decay table.
    for (int i = tid; i < T_DIM * F_DIM; i += 320) sW[i] = W[i];
    if (tid < T_DIM) sDec[tid] = __expf(-LAM * (float)tid);
    __syncthreads();

    const int wave = tid >> 5;
    const int lane = tid & 31;

    const int mT = blockIdx.x * 10 + wave;   // 625 * 10 = 6250 M-tiles
    const int halfSel = lane >> 4;           // 0: lanes 0-15, 1: lanes 16-31
    const int koff    = halfSel * 2;         // K sub-offset for this half-wave
    const int row     = mT * 16 + (lane & 15);
    const int nLane   = lane & 15;

    const float* __restrict__ xrow = x + (long)row * T_DIM;

    v8f acc0 = {}, acc1 = {}, acc2 = {}, acc3 = {};
#pragma unroll 4
    for (int k = 0; k < T_DIM; k += 4) {
        const int ka = k + koff;
        // A operand (shared by all 4 N-tiles): x[row, ka..ka+1] * decay.
        v2f a;
        a.x = xrow[ka]     * sDec[ka];
        a.y = xrow[ka + 1] * sDec[ka + 1];
        // B operands for the 4 N-tiles from LDS.
        const int r0 = ka * F_DIM + nLane;
        const int r1 = (ka + 1) * F_DIM + nLane;
        v2f b0, b1, b2, b3;
        b0.x = sW[r0];      b0.y = sW[r1];
        b1.x = sW[r0 + 16]; b1.y = sW[r1 + 16];
        b2.x = sW[r0 + 32]; b2.y = sW[r1 + 32];
        b3.x = sW[r0 + 48]; b3.y = sW[r1 + 48];
        // 4 independent accumulators: no D->A/B WMMA hazard.
        acc0 = __builtin_amdgcn_wmma_f32_16x16x4_f32(false, a, false, b0,
                                                     (short)0, acc0, false, false);
        acc1 = __builtin_amdgcn_wmma_f32_16x16x4_f32(false, a, false, b1,
                                                     (short)0, acc1, false, false);
        acc2 = __builtin_amdgcn_wmma_f32_16x16x4_f32(false, a, false, b2,
                                                     (short)0, acc2, false, false);
        acc3 = __builtin_amdgcn_wmma_f32_16x16x4_f32(false, a, false, b3,
                                                     (short)0, acc3, false, false);
    }

    // Scatter the 16x64 stripe: VGPR r -> row (mT*16 + r + halfSel*8).
    const int mBase = mT * 16 + halfSel * 8;
#pragma unroll
    for (int r = 0; r < 8; ++r) {
        const long base = (long)(mBase + r) * F_DIM + nLane;
        h[base]      = acc0[r];
        h[base + 16] = acc1[r];
        h[base + 32] = acc2[r];
        h[base + 48] = acc3[r];
    }
}

// ---------------------------------------------------------------------------
// Kernel 2: per-node top-8 of 32 edge weights + weighted gather of h rows.
// One wave32 per node; lane l owns edge l. Rank by shuffle-compare (ties ->
// lower index wins, matching jax.lax.top_k). Selected (w, nbr) staged in LDS,
// then all 32 lanes accumulate 2 features each. h gathers hit the 192MB L2.
// ---------------------------------------------------------------------------
__global__ void __launch_bounds__(256)
gat_topk_gather_kernel(const float* __restrict__ ew,
                       const int*   __restrict__ nbr,
                       const float* __restrict__ h,
                       float* __restrict__ out)
{
    __shared__ float sw[8 * TOPK];
    __shared__ int   si[8 * TOPK];

    const int tid  = threadIdx.x;
    const int wave = tid >> 5;
    const int lane = tid & 31;
    const int node = blockIdx.x * 8 + wave;   // 100000 = 8 * 12500, all live

    const long ebase = (long)node * DEG;
    const float w = ew[ebase + lane];

    // Rank of this lane's weight among the 32 (descending, stable).
    int rank = 0;
#pragma unroll
    for (int j = 0; j < DEG; ++j) {
        const float wj = __shfl(w, j, 32);
        rank += (wj > w) || ((wj == w) && (j < lane));
    }
    if (rank < TOPK) {
        sw[wave * TOPK + rank] = w;
        si[wave * TOPK + rank] = nbr[ebase + lane];
    }
    __syncthreads();

    // Accumulate: each lane owns features (lane) and (lane+32).
    float acc0 = 0.0f, acc1 = 0.0f;
#pragma unroll
    for (int k = 0; k < TOPK; ++k) {
        const float wk = sw[wave * TOPK + k];
        const long  hb = (long)si[wave * TOPK + k] * F_DIM;
        acc0 += wk * h[hb + lane];
        acc1 += wk * h[hb + 32 + lane];
    }

    const long ob = (long)node * F_DIM;
    out[ob + lane]      = acc0;
    out[ob + 32 + lane] = acc1;
}

// ---------------------------------------------------------------------------
// Launch: inputs are {x, edge_index, edge_weight, W, a}. 'a' is mathematically
// dead (softmax over singleton axis == 1 -> attention == topw). h lives in
// d_ws (needs 100000*64*4 = 25.6 MB).
// ---------------------------------------------------------------------------
extern "C" void kernel_launch(void* const* d_in, const int* in_sizes, int n_in,
                              void* d_out, int out_size, void* d_ws, size_t ws_size,
                              hipStream_t stream)
{
    const float* x          = (const float*)d_in[0];
    const int*   edge_index = (const int*)  d_in[1];  // (2, N*DEG) flat
    const float* ew         = (const float*)d_in[2];
    const float* W          = (const float*)d_in[3];
    // d_in[4] == a : unused (see note above)

    float* out = (float*)d_out;
    float* h   = (float*)d_ws;                        // N*F fp32 scratch

    const int* dst = edge_index + (long)N_NODES * DEG; // edge_index[1]

    // 6250 M-stripes / 10 waves-per-block = 625 blocks exactly.
    gat_gemm_h_kernel<<<625, 320, 0, stream>>>(x, W, h);
    // 100000 nodes / 8 waves-per-block = 12500 blocks exactly.
    gat_topk_gather_kernel<<<12500, 256, 0, stream>>>(ew, dst, h, out);
}